// VectorQuantizer_87711822118965
// MI455X (gfx1250) — compile-verified
//
#include <hip/hip_runtime.h>
#include <hip/hip_bf16.h>
#include <cstdint>

// ---------------------------------------------------------------------------
// VQ-VAE vector quantizer for MI455X (gfx1250, wave32, WMMA + async-to-LDS).
// z: [32,256,32,32] f32, emb: [4096,256] f32.
// d_out (f32): [8388608] z_q | loss | commitment | codebook | [32768] idx(int32 bits)
// ---------------------------------------------------------------------------

typedef __attribute__((ext_vector_type(2))) float v2f;
typedef __attribute__((ext_vector_type(8))) float v8f;
typedef __attribute__((ext_vector_type(4))) int   v4i;

typedef __attribute__((address_space(1))) v4i* as1_v4i;
typedef __attribute__((address_space(3))) v4i* as3_v4i;

#define NTOK   32768   // 32*32*32 tokens
#define CDIM   256
#define KCODE  4096
#define HW     1024    // 32*32
#define ROWS_PER_BLOCK 128
#define CHUNK  16      // codes per LDS chunk (double buffered)
#define NCHUNK (KCODE / CHUNK)
#define BPAD   260     // 256 + 4 pad: 1040B row stride = 16B aligned, banks (4t+k)%64
#define NELEM  8388608 // 32*256*32*32

#if defined(__HIP_DEVICE_COMPILE__) && \
    __has_builtin(__builtin_amdgcn_global_load_async_to_lds_b128) && \
    __has_builtin(__builtin_amdgcn_s_wait_asynccnt)
#define VQ_ASYNC 1
#else
#define VQ_ASYNC 0
#endif

// ---------------------------------------------------------------------------
// Kernel 1: enorm[k] = sum_c emb[k][c]^2
// ---------------------------------------------------------------------------
__global__ __launch_bounds__(256) void vq_enorm(const float* __restrict__ emb,
                                                float* __restrict__ enorm) {
    __shared__ float red[256];
    const int tid  = threadIdx.x;
    const int sub  = tid >> 4;          // code within block (0..15)
    const int lane = tid & 15;          // 16 lanes per code
    const int code = blockIdx.x * 16 + sub;

    float s = 0.f;
    const float* row = emb + (size_t)code * CDIM;
    #pragma unroll
    for (int c = lane; c < CDIM; c += 16) {
        float v = row[c];
        s += v * v;
    }
    red[tid] = s;
    __syncthreads();
    #pragma unroll
    for (int off = 8; off > 0; off >>= 1) {
        if (lane < off) red[tid] += red[tid + off];
        __syncthreads();
    }
    if (lane == 0) enorm[code] = red[tid];
}

// ---------------------------------------------------------------------------
// K-split dual-chain WMMA tile: acc0 <- k[0,128), acc1 <- k[128,256)
// ---------------------------------------------------------------------------
__device__ __forceinline__ void vq_compute_tile(const float* __restrict__ zb,
                                                const float (* __restrict__ B)[BPAD],
                                                int t, int g, v8f& acc0, v8f& acc1) {
    #pragma unroll 8
    for (int k0 = 0; k0 < 128; k0 += 4) {
        const int k = k0 + 2 * g;
        v2f a0, a1, b0, b1;
        a0.x = zb[(size_t)k * HW];           a0.y = zb[(size_t)(k + 1) * HW];
        a1.x = zb[(size_t)(k + 128) * HW];   a1.y = zb[(size_t)(k + 129) * HW];
        b0.x = B[t][k];                      b0.y = B[t][k + 1];
        b1.x = B[t][k + 128];                b1.y = B[t][k + 129];
        acc0 = __builtin_amdgcn_wmma_f32_16x16x4_f32(false, a0, false, b0,
                                                     (short)0, acc0, false, false);
        acc1 = __builtin_amdgcn_wmma_f32_16x16x4_f32(false, a1, false, b1,
                                                     (short)0, acc1, false, false);
    }
}

#if VQ_ASYNC
// Issue one 16-row (16KB) codebook chunk as 4 async B128 loads per thread.
__device__ __forceinline__ void vq_issue_chunk(const float* __restrict__ emb,
                                               float (*buf)[BPAD], int cBase, int tid) {
    #pragma unroll
    for (int j = 0; j < 4; ++j) {
        const int id = tid + 256 * j;       // 1024 x 16B chunks
        const int r  = id >> 6;             // codebook row 0..15
        const int c  = (id & 63) * 4;       // float column 0..252
        __builtin_amdgcn_global_load_async_to_lds_b128(
            (as1_v4i)(uintptr_t)(emb + (size_t)(cBase + r) * CDIM + c),
            (as3_v4i)(uintptr_t)(&buf[r][c]),
            0, 0);
    }
}
#endif

// ---------------------------------------------------------------------------
// Kernel 2: argmin_k ( enorm[k] - 2 * z_flat[n] . emb[k] ) via f32 WMMA.
// 256 blocks (128 rows each) x 256 threads (8 waves, 16-row strip per wave).
// B chunks double-buffered through LDS with async-to-LDS loads.
// ---------------------------------------------------------------------------
__global__ __launch_bounds__(256) void vq_argmin(const float* __restrict__ z,
                                                 const float* __restrict__ emb,
                                                 const float* __restrict__ enorm,
                                                 int* __restrict__ idxOut) {
    __shared__ float Bs[2][CHUNK][BPAD];     // 33,280 B
    __shared__ float bV[ROWS_PER_BLOCK][16]; //  8,192 B
    __shared__ int   bI[ROWS_PER_BLOCK][16]; //  8,192 B

    const int tid  = threadIdx.x;
    const int w    = tid >> 5;        // wave id 0..7
    const int lane = tid & 31;
    const int t    = lane & 15;       // A row / B col within tile
    const int g    = (lane >> 4) & 1; // K-group select

    const int rowBase = blockIdx.x * ROWS_PER_BLOCK;
    const int bIdx    = rowBase >> 10;                    // batch (const per block)
    const int hwA     = (rowBase & (HW - 1)) + w * 16 + t;
    // z[b][c][hw] : A element for row t, dim k -> zb[k * HW]
    const float* zb = z + ((size_t)bIdx * CDIM) * HW + hwA;

    float bestV[8];
    int   bestN[8];
    #pragma unroll
    for (int v = 0; v < 8; ++v) { bestV[v] = __builtin_inff(); bestN[v] = 0; }

#if VQ_ASYNC
    vq_issue_chunk(emb, Bs[0], 0, tid);
#endif

    for (int chunk = 0; chunk < NCHUNK; ++chunk) {
        const int cBase = chunk * CHUNK;
        __syncthreads();                         // next buffer free for overwrite
#if VQ_ASYNC
        if (chunk + 1 < NCHUNK) {
            vq_issue_chunk(emb, Bs[(chunk + 1) & 1], cBase + CHUNK, tid);
            __builtin_amdgcn_s_wait_asynccnt(4); // drain this chunk, keep next in flight
        } else {
            __builtin_amdgcn_s_wait_asynccnt(0);
        }
        __syncthreads();                         // chunk data visible to all waves
        const float (*B)[BPAD] = Bs[chunk & 1];
#else
        #pragma unroll 4
        for (int r = 0; r < CHUNK; ++r)
            Bs[0][r][tid] = emb[(size_t)(cBase + r) * CDIM + tid];
        __syncthreads();
        if (cBase + CHUNK < KCODE)
            __builtin_prefetch(&emb[(size_t)(cBase + CHUNK) * CDIM + tid], 0, 1);
        const float (*B)[BPAD] = Bs[0];
#endif

        v8f acc0 = {0.f,0.f,0.f,0.f,0.f,0.f,0.f,0.f};
        v8f acc1 = {0.f,0.f,0.f,0.f,0.f,0.f,0.f,0.f};
        vq_compute_tile(zb, B, t, g, acc0, acc1);

        // d = ||e||^2 - 2 * (z.e); per-lane n strictly increases with chunk,
        // so strict < keeps the earliest index (matches jnp.argmin ties).
        const int   n  = cBase + t;
        const float en = enorm[n];
        #pragma unroll
        for (int v = 0; v < 8; ++v) {
            const float d = en - 2.0f * (acc0[v] + acc1[v]);
            if (d < bestV[v]) { bestV[v] = d; bestN[v] = n; }
        }
    }

    // cross-lane finish: 16 candidate slots per row
    #pragma unroll
    for (int v = 0; v < 8; ++v) {
        const int r = w * 16 + v + 8 * g;   // D-matrix row m = v + 8*g
        bV[r][t] = bestV[v];
        bI[r][t] = bestN[v];
    }
    __syncthreads();

    if (tid < ROWS_PER_BLOCK) {
        float bv = bV[tid][0];
        int   bn = bI[tid][0];
        #pragma unroll
        for (int s = 1; s < 16; ++s) {
            float v = bV[tid][s];
            int   n = bI[tid][s];
            if (v < bv || (v == bv && n < bn)) { bv = v; bn = n; }
        }
        idxOut[rowBase + tid] = bn;
    }
}

// ---------------------------------------------------------------------------
// Kernel 3: z_q gather (in [b,c,h,w] order) + per-block sq-diff partial sums.
// 2048 blocks x 256 threads x 16 elems. Deterministic tree reduction.
// ---------------------------------------------------------------------------
__global__ __launch_bounds__(256) void vq_gather(const float* __restrict__ z,
                                                 const float* __restrict__ emb,
                                                 const int* __restrict__ idx,
                                                 float* __restrict__ out,
                                                 float* __restrict__ partial) {
    __shared__ float red[256];
    const int tid  = threadIdx.x;
    const int base = blockIdx.x * 4096 + tid;

    float acc = 0.f;
    #pragma unroll
    for (int j = 0; j < 16; ++j) {
        const int i  = base + j * 256;       // output/input linear index [b,c,h,w]
        const int hw = i & (HW - 1);
        const int c  = (i >> 10) & (CDIM - 1);
        const int b  = i >> 18;
        const int n  = (b << 10) | hw;       // flat token id
        const int k  = idx[n];
        const float v = emb[(size_t)k * CDIM + c];
        out[i] = v;                           // straight-through fwd value == z_q
        const float d = v - z[i];
        acc += d * d;
    }
    red[tid] = acc;
    __syncthreads();
    #pragma unroll
    for (int off = 128; off > 0; off >>= 1) {
        if (tid < off) red[tid] += red[tid + off];
        __syncthreads();
    }
    if (tid == 0) partial[blockIdx.x] = red[0];
}

// ---------------------------------------------------------------------------
// Kernel 4: finalize losses (deterministic fixed-order sum of 2048 partials).
// ---------------------------------------------------------------------------
__global__ __launch_bounds__(256) void vq_final(const float* __restrict__ partial,
                                                float* __restrict__ out) {
    __shared__ float red[256];
    const int tid = threadIdx.x;
    float s = 0.f;
    #pragma unroll
    for (int j = 0; j < 8; ++j) s += partial[tid + j * 256];
    red[tid] = s;
    __syncthreads();
    #pragma unroll
    for (int off = 128; off > 0; off >>= 1) {
        if (tid < off) red[tid] += red[tid + off];
        __syncthreads();
    }
    if (tid == 0) {
        const float mse = red[0] / (float)NELEM;
        out[NELEM + 0] = 1.25f * mse;  // loss = commitment + codebook
        out[NELEM + 1] = 0.25f * mse;  // commitment
        out[NELEM + 2] = mse;          // codebook
    }
}

// ---------------------------------------------------------------------------
extern "C" void kernel_launch(void* const* d_in, const int* in_sizes, int n_in,
                              void* d_out, int out_size, void* d_ws, size_t ws_size,
                              hipStream_t stream) {
    const float* z   = (const float*)d_in[0];   // [32,256,32,32]
    const float* emb = (const float*)d_in[1];   // [4096,256]
    float* out = (float*)d_out;

    float* enorm   = (float*)d_ws;              // 4096 f32
    float* partial = enorm + KCODE;             // 2048 f32
    int*   idxOut  = (int*)(out + NELEM + 3);   // indices region of d_out

    vq_enorm <<<KCODE / 16,            256, 0, stream>>>(emb, enorm);
    vq_argmin<<<NTOK / ROWS_PER_BLOCK, 256, 0, stream>>>(z, emb, enorm, idxOut);
    vq_gather<<<NELEM / 4096,          256, 0, stream>>>(z, emb, idxOut, out, partial);
    vq_final <<<1,                     256, 0, stream>>>(partial, out);
}